// PseudoEnergyTerm_18880676233905
// MI455X (gfx1250) — compile-verified
//
#include <hip/hip_runtime.h>

// Problem constants (match reference)
#define N_P   50000
#define N_L   50000
#define DD    256
#define K2    512     // 2*D
#define OUTC  256
#define N_E   800000
#define BN_EPS 1e-5f

typedef __attribute__((ext_vector_type(2))) float v2f;
typedef __attribute__((ext_vector_type(8))) float v8f;

// ---------------------------------------------------------------- zero (4-byte words)
__global__ void zero_i32(int* __restrict__ p, long long n) {
    long long i = (long long)blockIdx.x * blockDim.x + threadIdx.x;
    long long stride = (long long)gridDim.x * blockDim.x;
    for (; i < n; i += stride) p[i] = 0;
}

// ---------------------------------------------------------------- edge histogram (counting sort pass 1)
__global__ void hist_kernel(const int* __restrict__ erow, const int* __restrict__ ecol,
                            int* __restrict__ cnt_l, int* __restrict__ cnt_p) {
    int i = (int)(blockIdx.x * blockDim.x + threadIdx.x);
    int stride = (int)(gridDim.x * blockDim.x);
    for (; i < N_E; i += stride) {
        atomicAdd(&cnt_l[ecol[i]], 1);
        atomicAdd(&cnt_p[erow[i]], 1);
    }
}

// ---------------------------------------------------------------- single-block exclusive scan, in place
// data[0..n-1]: counts -> exclusive prefix; data[n] = total
__global__ void scan_kernel(int* __restrict__ data, int n) {
    __shared__ int tmp[256];
    __shared__ int carry;
    if (threadIdx.x == 0) carry = 0;
    __syncthreads();
    for (int base = 0; base < n; base += 256) {
        int i = base + (int)threadIdx.x;
        int v = (i < n) ? data[i] : 0;
        tmp[threadIdx.x] = v;
        __syncthreads();
        #pragma unroll
        for (int off = 1; off < 256; off <<= 1) {
            int t = (threadIdx.x >= (unsigned)off) ? tmp[threadIdx.x - off] : 0;
            __syncthreads();
            tmp[threadIdx.x] += t;
            __syncthreads();
        }
        int incl = tmp[threadIdx.x];
        if (i < n) data[i] = carry + (incl - v);
        __syncthreads();
        if (threadIdx.x == 255) carry += tmp[255];
        __syncthreads();
    }
    if (threadIdx.x == 0) data[n] = carry;
}

// ---------------------------------------------------------------- counting sort pass 2: scatter edge ids
__global__ void scatter_kernel(const int* __restrict__ erow, const int* __restrict__ ecol,
                               const int* __restrict__ start_l, const int* __restrict__ start_p,
                               int* __restrict__ cur_l, int* __restrict__ cur_p,
                               int* __restrict__ perm_l, int* __restrict__ perm_p) {
    int i = (int)(blockIdx.x * blockDim.x + threadIdx.x);
    int stride = (int)(gridDim.x * blockDim.x);
    for (; i < N_E; i += stride) {
        int c = ecol[i];
        perm_l[start_l[c] + atomicAdd(&cur_l[c], 1)] = i;
        int r = erow[i];
        perm_p[start_p[r] + atomicAdd(&cur_p[r], 1)] = i;
    }
}

// ---------------------------------------------------------------- gather SpMM: one wave32 per destination node
// out[node] = sum over incident edges e: eval[e] * xsrc[srcidx[e]]
__global__ void spmm_gather_kernel(const int* __restrict__ start, const int* __restrict__ perm,
                                   const int* __restrict__ srcidx, const float* __restrict__ evals,
                                   const float* __restrict__ xsrc, float* __restrict__ out,
                                   int nnodes) {
    int wave = (int)((blockIdx.x * blockDim.x + threadIdx.x) >> 5);
    int lane = (int)(threadIdx.x & 31);
    int nw   = (int)((gridDim.x * blockDim.x) >> 5);
    for (int node = wave; node < nnodes; node += nw) {
        int s0 = start[node], s1 = start[node + 1];
        float acc[8] = {0.f, 0.f, 0.f, 0.f, 0.f, 0.f, 0.f, 0.f};
        for (int idx = s0; idx < s1; ++idx) {
            int e = perm[idx];
            int src = srcidx[e];
            float v = evals[e];
            const float* row = xsrc + (long long)src * DD;
            #pragma unroll
            for (int j = 0; j < 8; ++j) acc[j] += v * row[lane + 32 * j];
        }
        float* o = out + (long long)node * DD;
        #pragma unroll
        for (int j = 0; j < 8; ++j) o[lane + 32 * j] = acc[j];
    }
}

// ---------------------------------------------------------------- W [OUTC,K2] -> Wt [K2,OUTC]
__global__ void transpose_kernel(const float* __restrict__ W, float* __restrict__ Wt) {
    int i = (int)(blockIdx.x * blockDim.x + threadIdx.x);
    int stride = (int)(gridDim.x * blockDim.x);
    int total = OUTC * K2;
    for (; i < total; i += stride) {
        int k = i / OUTC, n = i - k * OUTC;   // i = k*OUTC + n (coalesced writes)
        Wt[i] = W[n * K2 + k];
    }
}

// ---------------------------------------------------------------- GEMM: h = concat(x1, feat) @ W^T + b
// Block = 8 waves = one 16-row M-tile x all 256 output cols.
// A slab (16 x 512 f32) staged once in LDS (stride 516 -> conflict-free ds_load_b64
// fragment reads: lane l<16 hits banks {4l+k0, 4l+k0+1}, lanes 16-31 the interleaved pair).
// Each wave computes two 16x16 tiles with V_WMMA_F32_16X16X4_F32.
#define A_STRIDE 516
__global__ __launch_bounds__(256) void gemm_kernel(const float* __restrict__ x1,
                                                   const float* __restrict__ feat,
                                                   const float* __restrict__ Wt,   // [K2, OUTC]
                                                   const float* __restrict__ bias,
                                                   float* __restrict__ h, int nrows) {
    __shared__ float As[16 * A_STRIDE];     // 33,024 B
    int tM  = (int)blockIdx.x;              // nrows/16 blocks, exact
    int tid = (int)threadIdx.x;

    // Cooperative A-slab load: thread t -> row t/16, columns [32*(t%16), +32)
    {
        int rrow = tid >> 4;
        int c0   = (tid & 15) * 32;
        long long grow = (long long)(tM * 16 + rrow);
        const float* src1 = x1   + grow * DD;
        const float* src2 = feat + grow * DD;
        float* dst = As + rrow * A_STRIDE;
        #pragma unroll
        for (int u = 0; u < 32; u += 4) {
            int c = c0 + u;                  // 32-aligned segments never straddle DD
            const float* s = (c < DD) ? (src1 + c) : (src2 + (c - DD));
            *(float4*)(dst + c) = *(const float4*)s;
        }
    }
    __syncthreads();

    int wv   = tid >> 5;
    int lane = tid & 31;
    int lrow = lane & 15;
    int half = lane >> 4;
    int n0 = wv * 32 + lrow;                // first sub-tile column
    int n1 = n0 + 16;                       // second sub-tile column
    const float* arow = As + lrow * A_STRIDE;

    v8f acc0 = {}, acc1 = {};
    for (int kb = 0; kb < K2; kb += 4) {
        int k0 = kb + half * 2;
        v2f A;  A[0]  = arow[k0];                    A[1]  = arow[k0 + 1];
        v2f B0; B0[0] = Wt[(kb + half) * OUTC + n0]; B0[1] = Wt[(kb + 2 + half) * OUTC + n0];
        v2f B1; B1[0] = Wt[(kb + half) * OUTC + n1]; B1[1] = Wt[(kb + 2 + half) * OUTC + n1];
        acc0 = __builtin_amdgcn_wmma_f32_16x16x4_f32(false, A, false, B0, (short)0, acc0, false, false);
        acc1 = __builtin_amdgcn_wmma_f32_16x16x4_f32(false, A, false, B1, (short)0, acc1, false, false);
    }

    float bv0 = bias[n0], bv1 = bias[n1];
    int mbase = tM * 16 + half * 8;
    #pragma unroll
    for (int i = 0; i < 8; ++i) {
        h[(long long)(mbase + i) * OUTC + n0] = acc0[i] + bv0;
        h[(long long)(mbase + i) * OUTC + n1] = acc1[i] + bv1;
    }
}

// ---------------------------------------------------------------- per-column sum / sum of squares
#define STAT_ROWS 128
__global__ void colstat_kernel(const float* __restrict__ h, int nrows,
                               float* __restrict__ sum, float* __restrict__ sq) {
    int col = (int)threadIdx.x;             // blockDim.x == OUTC
    int r0  = (int)blockIdx.x * STAT_ROWS;
    int r1  = r0 + STAT_ROWS; if (r1 > nrows) r1 = nrows;
    float s = 0.0f, q = 0.0f;
    for (int r = r0; r < r1; ++r) {
        float v = h[(long long)r * OUTC + col];
        s += v; q += v * v;
    }
    atomicAdd(&sum[col], s);
    atomicAdd(&sq[col],  q);
}

// ---------------------------------------------------------------- BN + affine + ReLU in place
__global__ void bn_relu_kernel(float* __restrict__ h, int nrows,
                               const float* __restrict__ sum, const float* __restrict__ sq,
                               const float* __restrict__ gamma, const float* __restrict__ beta) {
    long long n = (long long)nrows * OUTC;
    long long i = (long long)blockIdx.x * blockDim.x + threadIdx.x;
    long long stride = (long long)gridDim.x * blockDim.x;
    float invN = 1.0f / (float)nrows;
    for (; i < n; i += stride) {
        int col = (int)(i & (OUTC - 1));
        float mu  = sum[col] * invN;
        float var = fmaxf(sq[col] * invN - mu * mu, 0.0f);
        float inv = rsqrtf(var + BN_EPS);
        float v   = (h[i] - mu) * inv * gamma[col] + beta[col];
        h[i] = fmaxf(v, 0.0f);
    }
}

// ---------------------------------------------------------------- launch
extern "C" void kernel_launch(void* const* d_in, const int* in_sizes, int n_in,
                              void* d_out, int out_size, void* d_ws, size_t ws_size,
                              hipStream_t stream) {
    const float* px    = (const float*)d_in[0];
    const float* lx    = (const float*)d_in[1];
    const int*   erow  = (const int*)  d_in[2];
    const int*   ecol  = (const int*)  d_in[3];
    const float* eval  = (const float*)d_in[4];
    const float* W_pl  = (const float*)d_in[5];
    const float* b_pl  = (const float*)d_in[6];
    const float* g_pl  = (const float*)d_in[7];
    const float* be_pl = (const float*)d_in[8];
    const float* W_lp  = (const float*)d_in[9];
    const float* b_lp  = (const float*)d_in[10];
    const float* g_lp  = (const float*)d_in[11];
    const float* be_lp = (const float*)d_in[12];

    float* out  = (float*)d_out;
    float* px_p = out;                               // [N_L, OUTC]
    float* lx_p = out + (long long)N_L * OUTC;       // [N_P, OUTC]

    // Workspace layout (all 4-byte aligned)
    float* ws    = (float*)d_ws;
    float* x1_l  = ws;                               // N_L*DD
    float* x1_p  = x1_l + (long long)N_L * DD;       // N_P*DD
    float* Wt_lp = x1_p + (long long)N_P * DD;       // K2*OUTC
    float* Wt_pl = Wt_lp + (long long)K2 * OUTC;     // K2*OUTC
    float* stats = Wt_pl + (long long)K2 * OUTC;     // 4*OUTC  (zeroed region starts here)
    float* sum_l = stats;
    float* sq_l  = stats + OUTC;
    float* sum_p = stats + 2 * OUTC;
    float* sq_p  = stats + 3 * OUTC;
    int* start_l = (int*)(stats + 4 * OUTC);         // N_L+1 (counts -> offsets)
    int* start_p = start_l + (N_L + 1);              // N_P+1
    int* cur_l   = start_p + (N_P + 1);              // N_L
    int* cur_p   = cur_l + N_L;                      // N_P
    int* perm_l  = cur_p + N_P;                      // N_E
    int* perm_p  = perm_l + N_E;                     // N_E

    // 1) zero stats + counts + cursors (contiguous 4-byte region)
    long long zwords = 4LL * OUTC + (N_L + 1) + (N_P + 1) + N_L + N_P;
    zero_i32<<<512, 256, 0, stream>>>((int*)stats, zwords);

    // 2) COO -> CSR counting sort (integer atomics only)
    hist_kernel<<<2048, 256, 0, stream>>>(erow, ecol, start_l, start_p);
    scan_kernel<<<1, 256, 0, stream>>>(start_l, N_L);
    scan_kernel<<<1, 256, 0, stream>>>(start_p, N_P);
    scatter_kernel<<<2048, 256, 0, stream>>>(erow, ecol, start_l, start_p,
                                             cur_l, cur_p, perm_l, perm_p);

    // 3) gather SpMM (no float atomics; write-once outputs)
    spmm_gather_kernel<<<6250, 256, 0, stream>>>(start_l, perm_l, erow, eval, px, x1_l, N_L);
    spmm_gather_kernel<<<6250, 256, 0, stream>>>(start_p, perm_p, ecol, eval, lx, x1_p, N_P);

    // 4) transpose weights for lane-contiguous B fragments
    transpose_kernel<<<512, 256, 0, stream>>>(W_lp, Wt_lp);
    transpose_kernel<<<512, 256, 0, stream>>>(W_pl, Wt_pl);

    // 5) WMMA GEMMs (one block per 16-row tile; 50000/16 = 3125 blocks, exact)
    gemm_kernel<<<N_L / 16, 256, 0, stream>>>(x1_l, lx, Wt_lp, b_lp, px_p, N_L);
    gemm_kernel<<<N_P / 16, 256, 0, stream>>>(x1_p, px, Wt_pl, b_pl, lx_p, N_P);

    // 6) BN statistics + apply
    colstat_kernel<<<(N_L + STAT_ROWS - 1) / STAT_ROWS, OUTC, 0, stream>>>(px_p, N_L, sum_l, sq_l);
    colstat_kernel<<<(N_P + STAT_ROWS - 1) / STAT_ROWS, OUTC, 0, stream>>>(lx_p, N_P, sum_p, sq_p);
    bn_relu_kernel<<<2048, 256, 0, stream>>>(px_p, N_L, sum_l, sq_l, g_lp, be_lp);
    bn_relu_kernel<<<2048, 256, 0, stream>>>(lx_p, N_P, sum_p, sq_p, g_pl, be_pl);
}